// TARelationConv_39513699123493
// MI455X (gfx1250) — compile-verified
//
#include <hip/hip_runtime.h>
#include <hip/hip_bf16.h>
#include <float.h>

// ---------------------------------------------------------------------------
// TARelationConv for MI455X (gfx1250, wave32).
// fp32 end-to-end using V_WMMA_F32_16X16X4_F32 (exact fp32 matrix math).
// Problem: S=64, N=256, L=40, OD=128, k+1=17 neighbors (k input is fixed 16).
// ~16 GFLOP fp32, all data L2-resident -> matrix-pipe bound; intermediates
// (rel/fg/ins, 142MB each if materialized) are fused away per (s,n) block.
// Workspace use: idx (1.06MB) + lang (1.25MB) + f (8MB) ~= 10.9MB.
// ---------------------------------------------------------------------------

typedef float v2f __attribute__((ext_vector_type(2)));
typedef float v8f __attribute__((ext_vector_type(8)));

#define S_      64
#define N_      256
#define L_      40
#define LP_     48      // L padded to WMMA multiple
#define KNB     17      // k+1 neighbors
#define OD_     128
#define LANGID_ 256
#define PCID_   128

static __device__ inline v8f wmma4(v2f a, v2f b, v8f c) {
  // D = A(16x4 f32) * B(4x16 f32) + C(16x16 f32)
  return __builtin_amdgcn_wmma_f32_16x16x4_f32(false, a, false, b, (short)0, c,
                                               false, false);
}

// ---------------------------------------------------------------------------
// Kernel 1: exact kNN (17 smallest d2, lexicographic (d2, idx) tie-break to
// match jax.lax.top_k ordering semantics). One block per sentence.
// ---------------------------------------------------------------------------
__global__ void knn_kernel(const float* __restrict__ coord,
                           int* __restrict__ idx_out) {
  __shared__ float cx[N_], cy[N_], cz[N_];
  const int s = blockIdx.x;
  const int n = threadIdx.x;
  const float* cs = coord + (size_t)s * N_ * 3;
  cx[n] = cs[n * 3 + 0];
  cy[n] = cs[n * 3 + 1];
  cz[n] = cs[n * 3 + 2];
  __syncthreads();

  const float px = cx[n], py = cy[n], pz = cz[n];
  float pd = -1.0f;
  int   pi = -1;
  int* out = idx_out + ((size_t)s * N_ + n) * KNB;
  for (int j = 0; j < KNB; ++j) {
    float bd = FLT_MAX;
    int   bi = 0x7fffffff;
    for (int m = 0; m < N_; ++m) {
      float dx = cx[m] - px, dy = cy[m] - py, dz = cz[m] - pz;
      float d2 = dx * dx + dy * dy + dz * dz;
      bool gtPrev = (d2 > pd) || (d2 == pd && m > pi);
      bool ltBest = (d2 < bd) || (d2 == bd && m < bi);
      if (gtPrev && ltBest) { bd = d2; bi = m; }
    }
    out[j] = bi;
    pd = bd;
    pi = bi;
  }
}

// ---------------------------------------------------------------------------
// Kernel 2: lang MLP, [S*L, 256] -> relu -> [.,128] -> [.,128]. Tiny; VALU.
// One block (128 threads) per (s,l) row.
// ---------------------------------------------------------------------------
__global__ void lang_mlp_kernel(const float* __restrict__ lang_feat,
                                const float* __restrict__ w1,
                                const float* __restrict__ b1,
                                const float* __restrict__ w2,
                                const float* __restrict__ b2,
                                float* __restrict__ lang_out) {
  __shared__ float xs[LANGID_];
  __shared__ float hs[OD_];
  const int row = blockIdx.x;  // s*L + l
  const int t   = threadIdx.x; // 0..127
  const float* x = lang_feat + (size_t)row * LANGID_;
  xs[t]       = x[t];
  xs[t + 128] = x[t + 128];
  __syncthreads();
  float acc = b1[t];
  for (int k = 0; k < LANGID_; ++k) acc += xs[k] * w1[k * OD_ + t];
  hs[t] = fmaxf(acc, 0.0f);
  __syncthreads();
  float acc2 = b2[t];
  for (int k = 0; k < OD_; ++k) acc2 += hs[k] * w2[k * OD_ + t];
  lang_out[(size_t)row * OD_ + t] = acc2;
}

// ---------------------------------------------------------------------------
// Kernel 3: feat MLP with fp32 WMMA. Block = 16 rows x 128 cols of one
// sentence; 8 waves, wave w owns output col-tile [16w, 16w+16).
// ---------------------------------------------------------------------------
__global__ void feat_mlp_kernel(const float* __restrict__ feat,
                                const float* __restrict__ w1,
                                const float* __restrict__ b1,
                                const float* __restrict__ w2,
                                const float* __restrict__ b2,
                                float* __restrict__ f_out) {
  __shared__ float Asm[16 * PCID_];
  __shared__ float Hsm[16 * OD_];
  const int s    = blockIdx.x >> 4;
  const int mt   = blockIdx.x & 15;
  const int tid  = threadIdx.x;
  const int lane = tid & 31, wave = tid >> 5;
  const int half = lane >> 4, lr = lane & 15;

  const float* A = feat + ((size_t)s * N_ + mt * 16) * PCID_;
  for (int i = tid; i < 16 * PCID_; i += 256) Asm[i] = A[i];
  __syncthreads();

  const int n0 = wave * 16;
  v8f c = {};
  for (int k0 = 0; k0 < PCID_; k0 += 4) {
    const int ka = k0 + 2 * half;
    v2f a, b;
    a.x = Asm[lr * PCID_ + ka];
    a.y = Asm[lr * PCID_ + ka + 1];
    b.x = w1[ka * OD_ + n0 + lr];
    b.y = w1[(ka + 1) * OD_ + n0 + lr];
    c = wmma4(a, b, c);
  }
  {
    const float bb = b1[n0 + lr];
#pragma unroll
    for (int v = 0; v < 8; ++v)
      Hsm[(v + 8 * half) * OD_ + n0 + lr] = fmaxf(c[v] + bb, 0.0f);
  }
  __syncthreads();

  v8f c2 = {};
  for (int k0 = 0; k0 < OD_; k0 += 4) {
    const int ka = k0 + 2 * half;
    v2f a, b;
    a.x = Hsm[lr * OD_ + ka];
    a.y = Hsm[lr * OD_ + ka + 1];
    b.x = w2[ka * OD_ + n0 + lr];
    b.y = w2[(ka + 1) * OD_ + n0 + lr];
    c2 = wmma4(a, b, c2);
  }
  {
    const float bb = b2[n0 + lr];
    float* fo = f_out + ((size_t)s * N_ + mt * 16) * OD_;
#pragma unroll
    for (int v = 0; v < 8; ++v)
      fo[(v + 8 * half) * OD_ + n0 + lr] = c2[v] + bb;
  }
}

// ---------------------------------------------------------------------------
// Kernel 4: fused per-(s,n): gather fg, rel MLP (WMMA), attn logits (WMMA),
// softmax+mask+renorm, ins (WMMA), combine + score. 17 rows padded to 32.
// LDS ~57KB (hbuf reused for ins).
// ---------------------------------------------------------------------------
__global__ void fused_kernel(const float* __restrict__ coord,
                             const float* __restrict__ lang_mask,
                             const float* __restrict__ rel_w1,
                             const float* __restrict__ rel_b1,
                             const float* __restrict__ rel_w2,
                             const float* __restrict__ rel_b2,
                             const int*   __restrict__ idx_ws,
                             const float* __restrict__ lang_ws,
                             const float* __restrict__ f_ws,
                             float* __restrict__ out,
                             float* __restrict__ score) {
  __shared__ float fg[32 * OD_];      // gathered f rows, rows 17..31 = 0
  __shared__ float relin[32 * 16];    // rel MLP input, 10 cols padded to 16
  __shared__ float hbuf[32 * OD_];    // rel hidden; later reused as ins
  __shared__ float relm[32 * OD_];    // rel output
  __shared__ float attnsm[32 * LP_];  // attn logits -> weights
  __shared__ int   idxsm[KNB];
  __shared__ float red[OD_];

  const int s    = blockIdx.x >> 8;
  const int n    = blockIdx.x & 255;
  const int tid  = threadIdx.x;
  const int lane = tid & 31, wave = tid >> 5;
  const int half = lane >> 4, lr = lane & 15;

  if (tid < KNB) idxsm[tid] = idx_ws[((size_t)s * N_ + n) * KNB + tid];
  for (int i = tid; i < 32 * 16; i += 256) relin[i] = 0.f;
  __syncthreads();

  // gather fg rows (zero padding rows)
  for (int i = tid; i < 32 * OD_; i += 256) {
    const int j = i >> 7, d = i & 127;
    fg[i] = (j < KNB) ? f_ws[((size_t)s * N_ + idxsm[j]) * OD_ + d] : 0.f;
  }
  // rel input rows: [nb(3), ctr(3), nb-ctr(3), ||nb-ctr||]
  if (tid < KNB) {
    const int j = tid;
    const float* cs = coord + (size_t)s * N_ * 3;
    const int m = idxsm[j];
    const float nx = cs[m * 3 + 0], ny = cs[m * 3 + 1], nz = cs[m * 3 + 2];
    const float ox = cs[n * 3 + 0], oy = cs[n * 3 + 1], oz = cs[n * 3 + 2];
    const float rx = nx - ox, ry = ny - oy, rz = nz - oz;
    const float dist = sqrtf(rx * rx + ry * ry + rz * rz);
    float* ri = relin + j * 16;
    ri[0] = nx; ri[1] = ny; ri[2] = nz;
    ri[3] = ox; ri[4] = oy; ri[5] = oz;
    ri[6] = rx; ri[7] = ry; ri[8] = rz;
    ri[9] = dist;
  }
  __syncthreads();

  // rel stage 1: h = relu(relin @ rel_w1 + b1)   M=32 K=16(pad) N=128
  for (int job = wave; job < 16; job += 8) {
    const int r0 = (job & 1) * 16, c0 = (job >> 1) * 16;
    v8f c = {};
    for (int k0 = 0; k0 < 16; k0 += 4) {
      const int ka = k0 + 2 * half;
      v2f a, b;
      a.x = relin[(r0 + lr) * 16 + ka];
      a.y = relin[(r0 + lr) * 16 + ka + 1];
      const int kb0 = min(ka, 9), kb1 = min(ka + 1, 9);  // A cols >=10 are 0
      b.x = rel_w1[kb0 * OD_ + c0 + lr];
      b.y = rel_w1[kb1 * OD_ + c0 + lr];
      c = wmma4(a, b, c);
    }
    const float bb = rel_b1[c0 + lr];
#pragma unroll
    for (int v = 0; v < 8; ++v)
      hbuf[(r0 + v + 8 * half) * OD_ + c0 + lr] = fmaxf(c[v] + bb, 0.f);
  }
  __syncthreads();

  // rel stage 2: rel = h @ rel_w2 + b2   M=32 K=128 N=128
  for (int job = wave; job < 16; job += 8) {
    const int r0 = (job & 1) * 16, c0 = (job >> 1) * 16;
    v8f c = {};
    for (int k0 = 0; k0 < OD_; k0 += 4) {
      const int ka = k0 + 2 * half;
      v2f a, b;
      a.x = hbuf[(r0 + lr) * OD_ + ka];
      a.y = hbuf[(r0 + lr) * OD_ + ka + 1];
      b.x = rel_w2[ka * OD_ + c0 + lr];
      b.y = rel_w2[(ka + 1) * OD_ + c0 + lr];
      c = wmma4(a, b, c);
    }
    const float bb = rel_b2[c0 + lr];
#pragma unroll
    for (int v = 0; v < 8; ++v)
      relm[(r0 + v + 8 * half) * OD_ + c0 + lr] = c[v] + bb;
  }
  __syncthreads();  // hbuf free from here

  // attn logits = fg @ lang^T   M=32 K=128 N=48 -> 6 tile jobs (waves 0..5)
  if (wave < 6) {
    const int r0 = (wave & 1) * 16, c0 = (wave >> 1) * 16;
    const float* langS = lang_ws + (size_t)s * L_ * OD_;
    const int lclamp = min(c0 + lr, L_ - 1);  // cols >=40 overwritten later
    v8f c = {};
    for (int k0 = 0; k0 < OD_; k0 += 4) {
      const int ka = k0 + 2 * half;
      v2f a, b;
      a.x = fg[(r0 + lr) * OD_ + ka];
      a.y = fg[(r0 + lr) * OD_ + ka + 1];
      b.x = langS[lclamp * OD_ + ka];       // B[k][l] = lang[l][k]
      b.y = langS[lclamp * OD_ + ka + 1];
      c = wmma4(a, b, c);
    }
#pragma unroll
    for (int v = 0; v < 8; ++v)
      attnsm[(r0 + v + 8 * half) * LP_ + c0 + lr] = c[v];
  }
  __syncthreads();

  // softmax over l, * mask, renormalize (exactly as reference); zero pad
  if (tid < 32) {
    float* rowp = attnsm + tid * LP_;
    if (tid < KNB) {
      float mx = -FLT_MAX;
      for (int l = 0; l < L_; ++l) mx = fmaxf(mx, rowp[l]);
      float ssum = 0.f;
      for (int l = 0; l < L_; ++l) {
        const float e = expf(rowp[l] - mx);
        rowp[l] = e;
        ssum += e;
      }
      const float inv = 1.f / ssum;
      const float* mk = lang_mask + (size_t)s * L_;
      float s2 = 0.f;
      for (int l = 0; l < L_; ++l) {
        const float p = rowp[l] * inv * mk[l];
        rowp[l] = p;
        s2 += p;
      }
      const float inv2 = 1.f / (s2 + 1e-7f);
      for (int l = 0; l < L_; ++l) rowp[l] *= inv2;
      for (int l = L_; l < LP_; ++l) rowp[l] = 0.f;
    } else {
      for (int l = 0; l < LP_; ++l) rowp[l] = 0.f;
    }
  }
  __syncthreads();

  // ins = attn @ lang   M=32 K=48(pad) N=128  (into hbuf)
  for (int job = wave; job < 16; job += 8) {
    const int r0 = (job & 1) * 16, c0 = (job >> 1) * 16;
    const float* langS = lang_ws + (size_t)s * L_ * OD_;
    v8f c = {};
    for (int k0 = 0; k0 < LP_; k0 += 4) {
      const int ka  = k0 + 2 * half;
      const int kb0 = min(ka, L_ - 1), kb1 = min(ka + 1, L_ - 1);
      v2f a, b;
      a.x = attnsm[(r0 + lr) * LP_ + ka];
      a.y = attnsm[(r0 + lr) * LP_ + ka + 1];
      b.x = langS[kb0 * OD_ + c0 + lr];
      b.y = langS[kb1 * OD_ + c0 + lr];
      c = wmma4(a, b, c);
    }
#pragma unroll
    for (int v = 0; v < 8; ++v)
      hbuf[(r0 + v + 8 * half) * OD_ + c0 + lr] = c[v];
  }
  __syncthreads();

  // combine: out[d] = sum_j fg*ins*rel + f[s,n,d];  score = sum_d out[d]
  if (tid < OD_) {
    const int d = tid;
    float acc = 0.f;
    for (int j = 0; j < KNB; ++j)
      acc += fg[j * OD_ + d] * hbuf[j * OD_ + d] * relm[j * OD_ + d];
    acc += f_ws[((size_t)s * N_ + n) * OD_ + d];
    out[((size_t)s * N_ + n) * OD_ + d] = acc;
    red[d] = acc;
  }
  __syncthreads();
  for (int off = 64; off > 0; off >>= 1) {
    if (tid < off) red[tid] += red[tid + off];
    __syncthreads();
  }
  if (tid == 0) score[(size_t)s * N_ + n] = red[0];
}

// ---------------------------------------------------------------------------
extern "C" void kernel_launch(void* const* d_in, const int* in_sizes, int n_in,
                              void* d_out, int out_size, void* d_ws,
                              size_t ws_size, hipStream_t stream) {
  (void)in_sizes; (void)n_in; (void)out_size; (void)ws_size;
  const float* feat      = (const float*)d_in[0];
  const float* coord     = (const float*)d_in[1];
  const float* lang_feat = (const float*)d_in[2];
  const float* lang_mask = (const float*)d_in[3];
  const float* rel_w1    = (const float*)d_in[4];
  const float* rel_b1    = (const float*)d_in[5];
  const float* rel_w2    = (const float*)d_in[6];
  const float* rel_b2    = (const float*)d_in[7];
  const float* lang_w1   = (const float*)d_in[8];
  const float* lang_b1   = (const float*)d_in[9];
  const float* lang_w2   = (const float*)d_in[10];
  const float* lang_b2   = (const float*)d_in[11];
  const float* feat_w1   = (const float*)d_in[12];
  const float* feat_b1   = (const float*)d_in[13];
  const float* feat_w2   = (const float*)d_in[14];
  const float* feat_b2   = (const float*)d_in[15];
  // d_in[16] is k == 16 -> KNB = 17 compiled in.

  // workspace: idx (S*N*17 int) | lang (S*L*OD f32) | f (S*N*OD f32) = ~10.9MB
  char* ws = (char*)d_ws;
  int*   idx_ws  = (int*)ws;
  float* lang_ws = (float*)(ws + (size_t)S_ * N_ * KNB * 4);
  float* f_ws    = (float*)(ws + (size_t)S_ * N_ * KNB * 4 +
                                 (size_t)S_ * L_ * OD_ * 4);

  float* out   = (float*)d_out;
  float* score = out + (size_t)S_ * N_ * OD_;

  knn_kernel<<<S_, N_, 0, stream>>>(coord, idx_ws);
  lang_mlp_kernel<<<S_ * L_, 128, 0, stream>>>(lang_feat, lang_w1, lang_b1,
                                               lang_w2, lang_b2, lang_ws);
  feat_mlp_kernel<<<S_ * (N_ / 16), 256, 0, stream>>>(feat, feat_w1, feat_b1,
                                                      feat_w2, feat_b2, f_ws);
  fused_kernel<<<S_ * N_, 256, 0, stream>>>(coord, lang_mask, rel_w1, rel_b1,
                                            rel_w2, rel_b2, idx_ws, lang_ws,
                                            f_ws, out, score);
}